// MultiHeadAttention_78099685310506
// MI455X (gfx1250) — compile-verified
//
#include <hip/hip_runtime.h>
#include <hip/hip_bf16.h>

// MHA forward for B=2, S=2048, D=1024, H=16, HD=64, fp32, causal, RoPE.
// GEMMs on V_WMMA_F32_16X16X4_F32; double-buffered TDM (tensor_load_to_lds)
// staging overlapped with WMMA compute.

#define BB 2
#define SS 2048
#define DD 1024
#define HH 16
#define HD 64
#define NQKV (3 * DD)

typedef __attribute__((ext_vector_type(2))) float v2f;
typedef __attribute__((ext_vector_type(8))) float v8f;
typedef __attribute__((ext_vector_type(4))) unsigned int u32x4;
typedef __attribute__((ext_vector_type(4))) int i32x4;
typedef __attribute__((ext_vector_type(8))) int i32x8;

static __device__ __forceinline__ v8f wmma_f32(v2f a, v2f b, v8f c) {
  return __builtin_amdgcn_wmma_f32_16x16x4_f32(false, a, false, b, (short)0, c,
                                               false, false);
}

#if __has_builtin(__builtin_amdgcn_tensor_load_to_lds) && \
    __has_builtin(__builtin_amdgcn_s_wait_tensorcnt)
#define USE_TDM 1
#else
#define USE_TDM 0
#endif

#if USE_TDM
// Pad-interval codes (DWORDs before pad): 5 -> 64 DW, 6 -> 128 DW.
#define PADI_64 5u
#define PADI_128 6u
// Pad-amount code: 7 -> 8 DWORDs.
#define PADA_8 7u

// 2D tile load, 4-byte elements: d0 elems/row, d1 rows, row stride stride0
// (elems). LDS row pitch = d0 + 8 dwords via D# pad feature.
static __device__ __forceinline__ void tdm_load_2d(unsigned int lds_off,
                                                   const void* gptr,
                                                   unsigned int d0,
                                                   unsigned int d1,
                                                   unsigned long long stride0,
                                                   unsigned int padi) {
  const unsigned long long ga = (unsigned long long)(uintptr_t)gptr;
  u32x4 g0;
  g0[0] = 1u;                                       // count=1, user descriptor
  g0[1] = lds_off;                                  // lds_addr
  g0[2] = (unsigned int)ga;                         // global_addr[31:0]
  g0[3] = (unsigned int)(ga >> 32) | (2u << 30);    // global_addr hi | type=2
  i32x8 g1;
  g1[0] = (int)((2u << 16)            // data_size = 4B
                | (1u << 20)          // pad_enable
                | (padi << 22)        // pad_interval
                | (PADA_8 << 25));    // pad_amount = 8 DW
  g1[1] = (int)((d0 & 0xffffu) << 16);                 // tensor_dim0 lo
  g1[2] = (int)((d0 >> 16) | ((d1 & 0xffffu) << 16));  // dim0 hi | dim1 lo
  g1[3] = (int)((d1 >> 16) | ((d0 & 0xffffu) << 16));  // dim1 hi | tile_dim0
  g1[4] = (int)(d1 & 0xffffu);                         // tile_dim1
  g1[5] = (int)(unsigned int)stride0;                  // dim0_stride lo
  g1[6] = (int)((stride0 >> 32) & 0xffffu);            // dim0_stride hi
  g1[7] = 0;
  i32x4 z4 = {};
#if __clang_major__ >= 23
  i32x8 z8 = {};
  __builtin_amdgcn_tensor_load_to_lds(g0, g1, z4, z4, z8, 0);
#else
  __builtin_amdgcn_tensor_load_to_lds(g0, g1, z4, z4, 0);
#endif
}
#endif  // USE_TDM

// Register-blocked 2M x 2N WMMA micro-kernel over one 64-deep K chunk.
// A: 32x64 tile (pitch 72), B: 64x128 tile (pitch 136).
static __device__ __forceinline__ void gemm_chunk(const float (*A)[72],
                                                  const float (*B)[136],
                                                  int ln, int g, int nw,
                                                  v8f& c00, v8f& c01, v8f& c10,
                                                  v8f& c11) {
#pragma unroll
  for (int kk = 0; kk < 64; kk += 4) {
    v2f a0, a1, b0, b1;
    a0.x = A[ln][kk + 2 * g];
    a0.y = A[ln][kk + 2 * g + 1];
    a1.x = A[16 + ln][kk + 2 * g];
    a1.y = A[16 + ln][kk + 2 * g + 1];
    b0.x = B[kk + 2 * g][nw + ln];
    b0.y = B[kk + 2 * g + 1][nw + ln];
    b1.x = B[kk + 2 * g][nw + 16 + ln];
    b1.y = B[kk + 2 * g + 1][nw + 16 + ln];
    c00 = wmma_f32(a0, b0, c00);
    c01 = wmma_f32(a0, b1, c01);
    c10 = wmma_f32(a1, b0, c10);
    c11 = wmma_f32(a1, b1, c11);
  }
}

// -----------------------------------------------------------------------------
// Kernel 1: QKV projection. Block tile 32(M) x 128(N), 4 waves; each wave owns
// a 32x32 register block (4 WMMA tiles). Double-buffered TDM staging.
// -----------------------------------------------------------------------------
__global__ void qkv_gemm_kernel(const float* __restrict__ x,
                                const float* __restrict__ W,
                                float* __restrict__ qws,
                                float* __restrict__ kws,
                                float* __restrict__ vws) {
  __shared__ float As[2][32][72];
  __shared__ float Bs[2][64][136];

  const int lane = threadIdx.x & 31;
  const int wave = threadIdx.x >> 5;
  const int g    = lane >> 4;
  const int ln   = lane & 15;

  const int m0 = blockIdx.y * 32;
  const int n0 = blockIdx.x * 128;
  const int nw = 32 * wave;

  v8f c00 = {}, c01 = {}, c10 = {}, c11 = {};
  const int NKC = DD / 64;

#if USE_TDM
  if (wave == 0) {
    tdm_load_2d((unsigned int)(uintptr_t)&As[0][0][0],
                &x[(size_t)m0 * DD], 64, 32, DD, PADI_64);
    tdm_load_2d((unsigned int)(uintptr_t)&Bs[0][0][0],
                &W[n0], 128, 64, NQKV, PADI_128);
  }
  for (int kc = 0; kc < NKC; ++kc) {
    if (wave == 0) __builtin_amdgcn_s_wait_tensorcnt(0);
    __syncthreads();  // publish buffer kc&1; prior readers of other buf done
    if (wave == 0 && kc + 1 < NKC) {
      const int pn = (kc + 1) & 1;
      tdm_load_2d((unsigned int)(uintptr_t)&As[pn][0][0],
                  &x[(size_t)m0 * DD + (kc + 1) * 64], 64, 32, DD, PADI_64);
      tdm_load_2d((unsigned int)(uintptr_t)&Bs[pn][0][0],
                  &W[(size_t)(kc + 1) * 64 * NQKV + n0], 128, 64, NQKV,
                  PADI_128);
    }
    const int p = kc & 1;
    gemm_chunk(As[p], Bs[p], ln, g, nw, c00, c01, c10, c11);
  }
#else
  for (int kc = 0; kc < NKC; ++kc) {
    const int k0 = kc * 64;
    for (int i = threadIdx.x; i < 32 * 16; i += 128) {
      const int row = i >> 4, c4 = i & 15;
      *(float4*)&As[0][row][c4 * 4] =
          *(const float4*)&x[(size_t)(m0 + row) * DD + k0 + c4 * 4];
    }
    for (int i = threadIdx.x; i < 64 * 32; i += 128) {
      const int row = i >> 5, c4 = i & 31;
      *(float4*)&Bs[0][row][c4 * 4] =
          *(const float4*)&W[(size_t)(k0 + row) * NQKV + n0 + c4 * 4];
    }
    __syncthreads();
    gemm_chunk(As[0], Bs[0], ln, g, nw, c00, c01, c10, c11);
    __syncthreads();
  }
#endif

#pragma unroll
  for (int ms = 0; ms < 2; ++ms) {
#pragma unroll
    for (int t2 = 0; t2 < 2; ++t2) {
      const v8f c = ms ? (t2 ? c11 : c10) : (t2 ? c01 : c00);
#pragma unroll
      for (int i = 0; i < 8; ++i) {
        const int m   = i + 8 * g;
        const int row = m0 + 16 * ms + m;
        const int bb  = row >> 11;
        const int s   = row & (SS - 1);
        const int col = n0 + nw + 16 * t2 + ln;
        const int t   = col >> 10;
        const int rem = col & (DD - 1);
        const int h   = rem >> 6;
        const int hd  = rem & (HD - 1);
        float* dst = (t == 0) ? qws : (t == 1) ? kws : vws;
        dst[(((size_t)bb * HH + h) * SS + s) * HD + hd] = c[i];
      }
    }
  }
}

// -----------------------------------------------------------------------------
// Kernel 2: RoPE on q and k, layout [B,H,S,HD].
// -----------------------------------------------------------------------------
__global__ void rope_kernel(float* __restrict__ q, float* __restrict__ k) {
  const int idx = blockIdx.x * blockDim.x + threadIdx.x;
  const int total = BB * HH * SS * (HD / 2);
  if (idx >= total) return;
  const int i  = idx & 31;
  const int s  = (idx >> 5) & (SS - 1);
  const int bh = idx >> 16;

  const float inv = __expf(-(float)(2 * i) * (9.2103403719761836f / 64.0f));
  const float ang = (float)s * inv;
  float sn, cs;
  __sincosf(ang, &sn, &cs);

  const size_t base = ((size_t)bh * SS + s) * HD;
  const float q1 = q[base + i], q2 = q[base + i + 32];
  q[base + i]      = q1 * cs - q2 * sn;
  q[base + i + 32] = q2 * cs + q1 * sn;
  const float k1 = k[base + i], k2 = k[base + i + 32];
  k[base + i]      = k1 * cs - k2 * sn;
  k[base + i + 32] = k2 * cs + k1 * sn;
}

// -----------------------------------------------------------------------------
// Kernel 3: flash-style causal attention per (b, h, 64-row q block).
// K/V tiles double-buffered via TDM; P transposed through per-wave LDS buffer.
// -----------------------------------------------------------------------------
__global__ void attn_kernel(const float* __restrict__ q,
                            const float* __restrict__ k,
                            const float* __restrict__ v,
                            float* __restrict__ o) {
  __shared__ float Ks[2][16][72];
  __shared__ float Vs[2][16][72];
  __shared__ float Ps[4][16][17];

  const int lane = threadIdx.x & 31;
  const int wave = threadIdx.x >> 5;
  const int g    = lane >> 4;
  const int ln   = lane & 15;

  const int bb = blockIdx.z;
  const int h  = blockIdx.y;
  const int q0 = blockIdx.x * 64;
  const int qr = q0 + wave * 16;

  const size_t headoff = ((size_t)bb * HH + h) * SS;
  const float* Qrow = q + (headoff + qr + ln) * HD;

  v2f qa[16];
#pragma unroll
  for (int kk = 0; kk < 16; ++kk) {
    const int d = kk * 4 + 2 * g;
    qa[kk].x = Qrow[d];
    qa[kk].y = Qrow[d + 1];
  }

  float rm[8], rl[8];
  v8f acc[4];
#pragma unroll
  for (int i = 0; i < 8; ++i) { rm[i] = -3.0e38f; rl[i] = 0.0f; }
#pragma unroll
  for (int j = 0; j < 4; ++j) acc[j] = (v8f){};

  const int ktmax = (q0 + 63) >> 4;

#if USE_TDM
  if (wave == 0) {
    const size_t src0 = headoff * HD + (size_t)q0 * 0;  // kt = 0
    tdm_load_2d((unsigned int)(uintptr_t)&Ks[0][0][0], &k[headoff * HD], 64, 16,
                HD, PADI_64);
    tdm_load_2d((unsigned int)(uintptr_t)&Vs[0][0][0], &v[headoff * HD], 64, 16,
                HD, PADI_64);
    (void)src0;
  }
#endif

  for (int kt = 0; kt <= ktmax; ++kt) {
#if USE_TDM
    if (wave == 0) __builtin_amdgcn_s_wait_tensorcnt(0);
    __syncthreads();  // publish buffer kt&1; readers of other buf are done
    if (wave == 0 && kt < ktmax) {
      const int pn = (kt + 1) & 1;
      const size_t src = (headoff + (kt + 1) * 16) * HD;
      tdm_load_2d((unsigned int)(uintptr_t)&Ks[pn][0][0], &k[src], 64, 16, HD,
                  PADI_64);
      tdm_load_2d((unsigned int)(uintptr_t)&Vs[pn][0][0], &v[src], 64, 16, HD,
                  PADI_64);
    }
    const int p = kt & 1;
#else
    const int p = 0;
    for (int i = threadIdx.x; i < 16 * 16; i += 128) {
      const int row = i >> 4, c4 = i & 15;
      const size_t src = (headoff + kt * 16 + row) * HD + c4 * 4;
      *(float4*)&Ks[0][row][c4 * 4] = *(const float4*)&k[src];
      *(float4*)&Vs[0][row][c4 * 4] = *(const float4*)&v[src];
    }
    __syncthreads();
#endif

    // S = Q (16x64) * K^T
    v8f sc = {};
#pragma unroll
    for (int kk = 0; kk < 16; ++kk) {
      v2f b;
      const int d = kk * 4 + 2 * g;
      b.x = Ks[p][ln][d];
      b.y = Ks[p][ln][d + 1];
      sc = wmma_f32(qa[kk], b, sc);
    }

    // Scale + causal mask + online softmax.
    const int key = kt * 16 + ln;
#pragma unroll
    for (int i = 0; i < 8; ++i) {
      const int m    = i + 8 * g;
      const int qrow = qr + m;
      float sv = (key <= qrow) ? sc[i] * 0.125f : -3.0e38f;
      float red = sv;
      red = fmaxf(red, __shfl_xor(red, 1, 32));
      red = fmaxf(red, __shfl_xor(red, 2, 32));
      red = fmaxf(red, __shfl_xor(red, 4, 32));
      red = fmaxf(red, __shfl_xor(red, 8, 32));
      const float nm = fmaxf(rm[i], red);
      const float f  = __expf(rm[i] - nm);
      rm[i] = nm;
      const float pv = __expf(sv - nm);
      float ps = pv;
      ps += __shfl_xor(ps, 1, 32);
      ps += __shfl_xor(ps, 2, 32);
      ps += __shfl_xor(ps, 4, 32);
      ps += __shfl_xor(ps, 8, 32);
      rl[i] = rl[i] * f + ps;
#pragma unroll
      for (int j = 0; j < 4; ++j) acc[j][i] *= f;
      Ps[wave][m][ln] = pv;
    }

    v2f pa[4];
#pragma unroll
    for (int kk = 0; kk < 4; ++kk) {
      const int d = kk * 4 + 2 * g;
      pa[kk].x = Ps[wave][ln][d];
      pa[kk].y = Ps[wave][ln][d + 1];
    }

    // O += P (16x16) * V (16x64)
#pragma unroll
    for (int j = 0; j < 4; ++j) {
#pragma unroll
      for (int kk = 0; kk < 4; ++kk) {
        v2f b;
        const int r = kk * 4 + 2 * g;
        b.x = Vs[p][r][16 * j + ln];
        b.y = Vs[p][r + 1][16 * j + ln];
        acc[j] = wmma_f32(pa[kk], b, acc[j]);
      }
    }
#if !USE_TDM
    __syncthreads();
#endif
  }

#pragma unroll
  for (int j = 0; j < 4; ++j) {
#pragma unroll
    for (int i = 0; i < 8; ++i) {
      const int m = i + 8 * g;
      o[(headoff + qr + m) * HD + 16 * j + ln] = acc[j][i] / rl[i];
    }
  }
}

// -----------------------------------------------------------------------------
// Kernel 4: output projection. Block tile 32 x 128; A gathered from [B,H,S,HD]
// (each 64-wide K chunk sits in one head -> contiguous 2D TDM tile).
// -----------------------------------------------------------------------------
__global__ void out_gemm_kernel(const float* __restrict__ ows,
                                const float* __restrict__ W,
                                float* __restrict__ out) {
  __shared__ float As[2][32][72];
  __shared__ float Bs[2][64][136];

  const int lane = threadIdx.x & 31;
  const int wave = threadIdx.x >> 5;
  const int g    = lane >> 4;
  const int ln   = lane & 15;

  const int m0 = blockIdx.y * 32;
  const int n0 = blockIdx.x * 128;
  const int nw = 32 * wave;

  const int bb0 = m0 >> 11;
  const int s0  = m0 & (SS - 1);

  v8f c00 = {}, c01 = {}, c10 = {}, c11 = {};
  const int NKC = DD / 64;  // chunk kc <-> head kc

#if USE_TDM
  if (wave == 0) {
    tdm_load_2d((unsigned int)(uintptr_t)&As[0][0][0],
                &ows[(((size_t)bb0 * HH + 0) * SS + s0) * HD], 64, 32, HD,
                PADI_64);
    tdm_load_2d((unsigned int)(uintptr_t)&Bs[0][0][0],
                &W[n0], 128, 64, DD, PADI_128);
  }
  for (int kc = 0; kc < NKC; ++kc) {
    if (wave == 0) __builtin_amdgcn_s_wait_tensorcnt(0);
    __syncthreads();
    if (wave == 0 && kc + 1 < NKC) {
      const int pn = (kc + 1) & 1;
      tdm_load_2d((unsigned int)(uintptr_t)&As[pn][0][0],
                  &ows[(((size_t)bb0 * HH + (kc + 1)) * SS + s0) * HD], 64, 32,
                  HD, PADI_64);
      tdm_load_2d((unsigned int)(uintptr_t)&Bs[pn][0][0],
                  &W[(size_t)(kc + 1) * 64 * DD + n0], 128, 64, DD, PADI_128);
    }
    const int p = kc & 1;
    gemm_chunk(As[p], Bs[p], ln, g, nw, c00, c01, c10, c11);
  }
#else
  for (int kc = 0; kc < NKC; ++kc) {
    const int k0 = kc * 64;
    for (int i = threadIdx.x; i < 32 * 16; i += 128) {
      const int row = i >> 4, c4 = i & 15;
      *(float4*)&As[0][row][c4 * 4] = *(const float4*)&ows[
          (((size_t)bb0 * HH + kc) * SS + s0 + row) * HD + c4 * 4];
    }
    for (int i = threadIdx.x; i < 64 * 32; i += 128) {
      const int row = i >> 5, c4 = i & 31;
      *(float4*)&Bs[0][row][c4 * 4] =
          *(const float4*)&W[(size_t)(k0 + row) * DD + n0 + c4 * 4];
    }
    __syncthreads();
    gemm_chunk(As[0], Bs[0], ln, g, nw, c00, c01, c10, c11);
    __syncthreads();
  }
#endif

#pragma unroll
  for (int ms = 0; ms < 2; ++ms) {
#pragma unroll
    for (int t2 = 0; t2 < 2; ++t2) {
      const v8f c = ms ? (t2 ? c11 : c10) : (t2 ? c01 : c00);
#pragma unroll
      for (int i = 0; i < 8; ++i) {
        const int m   = i + 8 * g;
        const int row = m0 + 16 * ms + m;
        const int bb  = row >> 11;
        const int s   = row & (SS - 1);
        out[((size_t)bb * SS + s) * DD + n0 + nw + 16 * t2 + ln] = c[i];
      }
    }
  }
}

// -----------------------------------------------------------------------------
extern "C" void kernel_launch(void* const* d_in, const int* in_sizes, int n_in,
                              void* d_out, int out_size, void* d_ws,
                              size_t ws_size, hipStream_t stream) {
  const float* x    = (const float*)d_in[0];
  const float* Wqkv = (const float*)d_in[1];
  const float* Wout = (const float*)d_in[2];
  // d_in[3] = dense causal mask: handled analytically, ignored.

  const size_t per = (size_t)BB * HH * SS * HD;
  float* qws = (float*)d_ws;
  float* kws = qws + per;
  float* vws = kws + per;
  float* ows = vws + per;
  float* out = (float*)d_out;

  qkv_gemm_kernel<<<dim3(NQKV / 128, (BB * SS) / 32), 128, 0, stream>>>(
      x, Wqkv, qws, kws, vws);

  {
    const int total = BB * HH * SS * (HD / 2);
    rope_kernel<<<(total + 255) / 256, 256, 0, stream>>>(qws, kws);
  }

  attn_kernel<<<dim3(SS / 64, HH, BB), 128, 0, stream>>>(qws, kws, vws, ows);

  out_gemm_kernel<<<dim3(DD / 128, (BB * SS) / 32), 128, 0, stream>>>(
      ows, Wout, out);
}